// DETRLoss_74156905333205
// MI455X (gfx1250) — compile-verified
//
#include <hip/hip_runtime.h>
#include <math.h>

#define NUM_C 2048   // NUM_CLASSES + 1
#define Bn 64
#define Qn 100
#define Nn 16
#define ROWS (Bn * Qn)   // 6400

typedef __attribute__((ext_vector_type(2))) float v2f;
typedef __attribute__((ext_vector_type(8))) float v8f;

// ---------------------------------------------------------------------------
// Kernel 0: zero target_classes and the partial-sum scratch (ws is poisoned).
// ---------------------------------------------------------------------------
__global__ void k_init(int* __restrict__ tc, float* __restrict__ partials) {
    int i = blockIdx.x * blockDim.x + threadIdx.x;
    if (i < ROWS) tc[i] = 0;
    if (i < 64)   partials[i] = 0.0f;
}

// ---------------------------------------------------------------------------
// Kernel 1: per 16 query-rows: logZ via V_WMMA_F32_16X16X4_F32 (A = exp tile,
// B = ones -> D accumulates exact f32 row sums), then the [16 x N] cost tile.
// One wave (32 threads) per block; EXEC all-1s around every WMMA.
// ---------------------------------------------------------------------------
__global__ __launch_bounds__(32)
void k_logz_cost(const float* __restrict__ logits,
                 const float* __restrict__ pred_time,
                 const int*   __restrict__ labels,
                 const float* __restrict__ tstamp,
                 float* __restrict__ logZ,
                 float* __restrict__ cost)
{
    const int lane    = threadIdx.x;
    const int rowBase = blockIdx.x * 16;
    __shared__ float sh_max[16];
    __shared__ float sh_lz[16];

    // Pass A: row maxes (lane-strided coalesced reads + shfl-xor reduce).
    for (int r = 0; r < 16; ++r) {
        const float* rowp = logits + (size_t)(rowBase + r) * NUM_C;
        float m = -INFINITY;
        for (int c = lane; c < NUM_C; c += 32) m = fmaxf(m, rowp[c]);
        for (int off = 16; off > 0; off >>= 1) m = fmaxf(m, __shfl_xor(m, off));
        if (lane == 0) sh_max[r] = m;
    }
    __syncthreads();

    // Pass B: sum of exp via WMMA. ISA A-layout (16x4 f32):
    //   lanes 0-15 : M = lane,    a.x->K=0, a.y->K=1
    //   lanes16-31 : M = lane-16, a.x->K=2, a.y->K=3
    const int   myrow = lane & 15;
    const int   koff  = (lane >> 4) << 1;
    const float mymax = sh_max[myrow];
    const float* rowp = logits + (size_t)(rowBase + myrow) * NUM_C;

    v8f acc = {0.f, 0.f, 0.f, 0.f, 0.f, 0.f, 0.f, 0.f};
    v2f ones; ones.x = 1.0f; ones.y = 1.0f;   // B = ones(4x16): layout-invariant
    for (int kb = 0; kb < NUM_C; kb += 4) {
        v2f a;
        a.x = expf(rowp[kb + koff]     - mymax);
        a.y = expf(rowp[kb + koff + 1] - mymax);
        acc = __builtin_amdgcn_wmma_f32_16x16x4_f32(
                  false, a, false, ones, (short)0, acc, false, false);
    }

    // D layout: VGPR j @ lane 0 -> (M=j, N=0); VGPR j @ lane 16 -> (M=j+8, N=0)
    if (lane == 0 || lane == 16) {
        const int rb = (lane >> 4) << 3;
        for (int j = 0; j < 8; ++j) {
            float lz = sh_max[rb + j] + logf(acc[j]);
            sh_lz[rb + j] = lz;
            logZ[rowBase + rb + j] = lz;
        }
    }
    __syncthreads();

    // Cost tile: C[p][n] = -softmax(logits)[label] + 2*|pt - ts|, nan_to_num.
    for (int idx = lane; idx < 16 * Nn; idx += 32) {
        int r = idx >> 4;
        int n = idx & 15;
        int p = rowBase + r;
        int b = p / Qn;
        int lab   = labels[b * Nn + n];
        float cc  = -expf(logits[(size_t)p * NUM_C + lab] - sh_lz[r]);
        float ctv = fabsf(pred_time[p] - tstamp[b * Nn + n]);
        float c   = cc + 2.0f * ctv;
        if (isnan(c))      c = 100.0f;
        else if (isinf(c)) c = (c > 0.0f) ? 100.0f : -100.0f;
        cost[(size_t)p * Nn + n] = c;
    }
}

// ---------------------------------------------------------------------------
// Kernel 2: Jonker-Volgenant per batch (exact port of the reference, f64
// duals to match numpy). Wave loads 16x100 transposed cost into LDS; lane 0
// solves. Writes matched query per target (qs) and scatters target_classes.
// ---------------------------------------------------------------------------
__global__ __launch_bounds__(32)
void k_hungarian(const float* __restrict__ cost,
                 const int*   __restrict__ labels,
                 int* __restrict__ qs,
                 int* __restrict__ tc)
{
    const int b = blockIdx.x;
    __shared__ float ct[Nn * Qn];   // ct[n][q] = C[b][q][n]
    for (int i = threadIdx.x; i < Nn * Qn; i += 32) {
        int n = i / Qn, q = i % Qn;
        ct[i] = cost[(size_t)b * Qn * Nn + q * Nn + n];
    }
    __syncthreads();

    if (threadIdx.x == 0) {
        double u[Nn + 1], v[Qn + 1], minv[Qn + 1];
        int    p[Qn + 1], way[Qn + 1];
        bool   used[Qn + 1];
        for (int i = 0; i <= Nn; ++i) u[i] = 0.0;
        for (int j = 0; j <= Qn; ++j) { v[j] = 0.0; p[j] = 0; }

        for (int i = 1; i <= Nn; ++i) {
            p[0] = i;
            int j0 = 0;
            for (int j = 0; j <= Qn; ++j) { minv[j] = INFINITY; used[j] = false; way[j] = 0; }
            for (;;) {
                used[j0] = true;
                int i0 = p[j0];
                double delta = INFINITY; int j1 = -1;
                for (int j = 1; j <= Qn; ++j) {
                    if (!used[j]) {
                        double cur = (double)ct[(i0 - 1) * Qn + (j - 1)] - u[i0] - v[j];
                        if (cur < minv[j]) { minv[j] = cur; way[j] = j0; }
                        if (minv[j] < delta) { delta = minv[j]; j1 = j; }  // first-min tie-break
                    }
                }
                for (int j = 0; j <= Qn; ++j) {
                    if (used[j]) { u[p[j]] += delta; v[j] -= delta; }
                    else         minv[j] -= delta;
                }
                j0 = j1;
                if (p[j0] == 0) break;
            }
            while (j0 != 0) { int jn = way[j0]; p[j0] = p[jn]; j0 = jn; }
        }
        for (int j = 1; j <= Qn; ++j) {
            if (p[j] > 0) {
                int row = p[j] - 1;   // target index
                int col = j - 1;      // matched query index
                qs[b * Nn + row] = col;
                tc[b * Qn + col] = labels[b * Nn + row];
            }
        }
    }
}

// ---------------------------------------------------------------------------
// Kernel 3: weighted CE partial sums. 25 blocks x 256 = 6400 rows exactly.
// Deterministic fixed-order LDS tree reduction -> partials[2*blk], [2*blk+1].
// ---------------------------------------------------------------------------
__global__ __launch_bounds__(256)
void k_ce(const float* __restrict__ logits,
          const float* __restrict__ logZ,
          const int*   __restrict__ tc,
          float* __restrict__ partials)
{
    __shared__ float s0[256], s1[256];
    const int tid = threadIdx.x;
    const int p   = blockIdx.x * 256 + tid;
    int   t   = tc[p];
    float w   = (t == 0) ? 0.1f : 1.0f;
    float nll = logZ[p] - logits[(size_t)p * NUM_C + t];
    s0[tid] = w * nll;
    s1[tid] = w;
    __syncthreads();
    for (int off = 128; off > 0; off >>= 1) {
        if (tid < off) { s0[tid] += s0[tid + off]; s1[tid] += s1[tid + off]; }
        __syncthreads();
    }
    if (tid == 0) {
        partials[blockIdx.x * 2]     = s0[0];
        partials[blockIdx.x * 2 + 1] = s1[0];
    }
}

// ---------------------------------------------------------------------------
// Kernel 4: L1 time-loss partial sum over the 1024 matched pairs (1 block).
// ---------------------------------------------------------------------------
__global__ __launch_bounds__(256)
void k_time(const float* __restrict__ pred_time,
            const float* __restrict__ tstamp,
            const int*   __restrict__ qs,
            float* __restrict__ partials)
{
    __shared__ float s[256];
    const int tid = threadIdx.x;
    float acc = 0.0f;
    for (int i = tid; i < Bn * Nn; i += 256) {
        int b = i >> 4;
        int q = qs[i];
        acc += fabsf(pred_time[b * Qn + q] - tstamp[i]);
    }
    s[tid] = acc;
    __syncthreads();
    for (int off = 128; off > 0; off >>= 1) {
        if (tid < off) s[tid] += s[tid + off];
        __syncthreads();
    }
    if (tid == 0) partials[50] = s[0];
}

// ---------------------------------------------------------------------------
// Kernel 5: final scalar combine (single thread, fixed order).
// ---------------------------------------------------------------------------
__global__ void k_final(const float* __restrict__ partials, float* __restrict__ out) {
    if (blockIdx.x == 0 && threadIdx.x == 0) {
        float wn = 0.0f, ws = 0.0f;
        for (int k = 0; k < 25; ++k) { wn += partials[2 * k]; ws += partials[2 * k + 1]; }
        float tsum = partials[50];
        out[0] = wn / ws + 2.0f * (tsum / (float)(Bn * Nn));
    }
}

// ---------------------------------------------------------------------------
extern "C" void kernel_launch(void* const* d_in, const int* in_sizes, int n_in,
                              void* d_out, int out_size, void* d_ws, size_t ws_size,
                              hipStream_t stream) {
    (void)in_sizes; (void)n_in; (void)out_size; (void)ws_size;
    const float* logits    = (const float*)d_in[0];  // [B,Q,2048]
    const float* pred_time = (const float*)d_in[1];  // [B,Q,1]
    const int*   labels    = (const int*)  d_in[2];  // [B,N]
    const float* tstamp    = (const float*)d_in[3];  // [B,N,1]
    float*       out       = (float*)d_out;

    char* ws = (char*)d_ws;
    float* cost     = (float*)ws;                 // 102400 f32 (409600 B)
    float* logZ     = (float*)(ws + 409600);      //   6400 f32
    int*   tc       = (int*)  (ws + 435200);      //   6400 i32 (target_classes)
    int*   qs       = (int*)  (ws + 460800);      //   1024 i32 (matched query idx)
    float* partials = (float*)(ws + 464896);      //     64 f32

    k_init     <<<26,       256, 0, stream>>>(tc, partials);
    k_logz_cost<<<ROWS / 16, 32, 0, stream>>>(logits, pred_time, labels, tstamp, logZ, cost);
    k_hungarian<<<Bn,        32, 0, stream>>>(cost, labels, qs, tc);
    k_ce       <<<25,       256, 0, stream>>>(logits, logZ, tc, partials);
    k_time     <<<1,        256, 0, stream>>>(pred_time, tstamp, qs, partials);
    k_final    <<<1,         32, 0, stream>>>(partials, out);
}